// GraphTripleConv_50740743635079
// MI455X (gfx1250) — compile-verified
//
#include <hip/hip_runtime.h>
#include <hip/hip_bf16.h>
#include <cstdint>
#include <cstddef>

// ---------------------------------------------------------------------------
// GraphTripleConv for MI455X (gfx1250, wave32, WMMA).
// - All GEMMs on V_WMMA_F32_16X16X4_F32 (exact fp32 tensor path).
// - A tiles staged via GLOBAL_LOAD_ASYNC_TO_LDS_B128 (ASYNCcnt), double-buffered.
// - B tiles staged transposed (Bs[n][k]) via load + ds_store so both WMMA
//   fragments are single ds_load_b64 per ISA 7.12.2 fp32 layouts.
// ---------------------------------------------------------------------------

typedef float v2f __attribute__((ext_vector_type(2)));
typedef float v4f __attribute__((ext_vector_type(4)));
typedef float v8f __attribute__((ext_vector_type(8)));

static constexpr int B_   = 32;
static constexpr int O_   = 256;
static constexpr int T_   = 1024;

static constexpr int BM = 128;
static constexpr int BN = 128;
static constexpr int BK = 32;
static constexpr int AKP = BK + 4;   // padded LDS row stride (floats); 144B = 16B-aligned
static constexpr int BKP = BK + 4;
static constexpr int NTHREADS = 256; // 8 waves (wave32)

__device__ __forceinline__ float sigmoidf_(float x) {
    return 1.0f / (1.0f + __expf(-x));
}

__device__ __forceinline__ v8f vzero8() {
    v8f z;
    #pragma unroll
    for (int e = 0; e < 8; ++e) z[e] = 0.0f;
    return z;
}

// Direct global -> LDS async copy (16B per lane), tracked by ASYNCcnt.
__device__ __forceinline__ void async_copy_b128(uint32_t lds_off, const float* src) {
    asm volatile("global_load_async_to_lds_b128 %0, %1, off"
                 :: "v"(lds_off), "v"(src)
                 : "memory");
}

__device__ __forceinline__ void async_wait0() {
#if __has_builtin(__builtin_amdgcn_s_wait_asynccnt)
    __builtin_amdgcn_s_wait_asynccnt(0);
#else
    asm volatile("s_wait_asynccnt 0x0" ::: "memory");
#endif
}

// D(M,N) = epilogue( A(M,K) @ W(K,N) + bias )
// AMODE 0: dense A.  AMODE 1: A rows gathered [obj[s] | pred | obj[o]] (K=1536).
// EMODE 0: relu -> out.  EMODE 1: relu, *conf, split {atomicAdd sums | new_p | atomicAdd sums}.
template <int AMODE, int EMODE>
__global__ __launch_bounds__(NTHREADS)
void gemm_wmma_f32(const float* __restrict__ A,
                   const float* __restrict__ objv,
                   const float* __restrict__ predv,
                   const int*   __restrict__ edges,     // (B*T, 2)
                   const float* __restrict__ W,         // K x N row-major
                   const float* __restrict__ bias,      // N
                   float* __restrict__ out,             // EMODE=0: M x N
                   float* __restrict__ sums,            // EMODE=1: (B*O, 512)
                   float* __restrict__ outP,            // EMODE=1: (B*T, 512)
                   const int* __restrict__ ttype,       // (B*T)
                   const int* __restrict__ pids,        // (B*T)
                   const unsigned char* __restrict__ mask8, // (B*T) bool
                   const float* __restrict__ ptw,       // (N_PRED_TYPES)
                   int M, int N, int K)
{
    __shared__ float As[2][BM * AKP];
    __shared__ float Bs[2][BN * BKP];

    const int tid  = threadIdx.x;
    const int lane = tid & 31;
    const int wave = tid >> 5;
    const int wm   = wave & 1;   // 2 wave-rows  -> 64 rows each
    const int wn   = wave >> 1;  // 4 wave-cols  -> 32 cols each

    const int m0 = blockIdx.y * BM;
    const int n0 = blockIdx.x * BN;

    v8f acc[4][2];
    #pragma unroll
    for (int i = 0; i < 4; ++i)
        #pragma unroll
        for (int j = 0; j < 2; ++j) acc[i][j] = vzero8();

    // ---- tile staging (A: async global->LDS, B: transposed via ds_store) ---
    auto stage_tile = [&](int k0, int buf) {
        #pragma unroll
        for (int it = 0; it < (BM * BK) / (NTHREADS * 4); ++it) {
            int p   = tid + it * NTHREADS;   // float4 index within A tile
            int row = p >> 3;                // BK/4 == 8 float4 per row
            int c4  = (p & 7) << 2;
            const float* src;
            if (AMODE == 0) {
                src = A + (size_t)(m0 + row) * K + k0 + c4;
            } else {
                int m = m0 + row;            // m = b*T + t
                int b = m >> 10;             // T_ == 1024
                if (k0 < 512) {
                    int s = edges[((size_t)m << 1) + 0];
                    src = objv + (size_t)(b * O_ + s) * 512 + (k0 + c4);
                } else if (k0 < 1024) {
                    src = predv + (size_t)m * 512 + (k0 - 512 + c4);
                } else {
                    int o = edges[((size_t)m << 1) + 1];
                    src = objv + (size_t)(b * O_ + o) * 512 + (k0 - 1024 + c4);
                }
            }
            uint32_t dst = (uint32_t)(uintptr_t)&As[buf][row * AKP + c4];
            async_copy_b128(dst, src);
        }
        #pragma unroll
        for (int it = 0; it < (BK * BN) / (NTHREADS * 4); ++it) {
            int p    = tid + it * NTHREADS;
            int krow = p >> 5;               // BN/4 == 32 float4 per k-row
            int c4   = (p & 31) << 2;        // n offset within tile
            v4f v = *(const v4f*)(W + (size_t)(k0 + krow) * N + n0 + c4);
            Bs[buf][(c4 + 0) * BKP + krow] = v.x;
            Bs[buf][(c4 + 1) * BKP + krow] = v.y;
            Bs[buf][(c4 + 2) * BKP + krow] = v.z;
            Bs[buf][(c4 + 3) * BKP + krow] = v.w;
        }
    };

    // ---- prologue: stage first tile ----------------------------------------
    stage_tile(0, 0);
    async_wait0();
    __syncthreads();

    const int ksteps = K / BK;
    for (int ks = 0; ks < ksteps; ++ks) {
        const int cur = ks & 1;

        // prefetch next tile into the other buffer while we compute
        if (ks + 1 < ksteps) stage_tile((ks + 1) * BK, cur ^ 1);

        // ---- WMMA inner loop: K in steps of 4 ------------------------------
        #pragma unroll
        for (int kk = 0; kk < BK; kk += 4) {
            // fp32 16x16x4 layout: lane L -> row/col (L&15), K pair 2*(L>>4)
            const int kf = kk + ((lane >> 4) << 1);
            v2f afr[4], bfr[2];
            #pragma unroll
            for (int i = 0; i < 4; ++i) {
                int m = wm * 64 + i * 16 + (lane & 15);
                afr[i] = *(const v2f*)&As[cur][m * AKP + kf];
            }
            #pragma unroll
            for (int j = 0; j < 2; ++j) {
                int n = wn * 32 + j * 16 + (lane & 15);
                bfr[j] = *(const v2f*)&Bs[cur][n * BKP + kf];
            }
            #pragma unroll
            for (int i = 0; i < 4; ++i)
                #pragma unroll
                for (int j = 0; j < 2; ++j)
                    acc[i][j] = __builtin_amdgcn_wmma_f32_16x16x4_f32(
                        false, afr[i], false, bfr[j],
                        (short)0, acc[i][j], false, false);
        }

        async_wait0();      // next tile's async A-loads have landed (own wave)
        __syncthreads();    // everyone's loads/stores visible; cur reusable
    }

    // ---- epilogue ----------------------------------------------------------
    #pragma unroll
    for (int j = 0; j < 2; ++j) {
        const int nloc = n0 + wn * 32 + j * 16 + (lane & 15);
        const float bv = bias[nloc];
        #pragma unroll
        for (int i = 0; i < 4; ++i) {
            #pragma unroll
            for (int r = 0; r < 8; ++r) {
                // C/D layout: VGPR r, lanes 0-15 -> M=r, lanes 16-31 -> M=r+8
                const int mloc = m0 + wm * 64 + i * 16 + r + ((lane >> 4) << 3);
                float v = acc[i][j][r] + bv;
                v = v > 0.0f ? v : 0.0f;
                if (EMODE == 0) {
                    out[(size_t)mloc * N + nloc] = v;
                } else {
                    const int tt = ttype[mloc];
                    const float conf =
                        (tt == 0) ? 1.0f : sigmoidf_(ptw[pids[mloc]]);
                    v *= conf;
                    const int b = mloc >> 10;   // mloc = b*T + t
                    if (nloc < 512) {
                        const int s = edges[((size_t)mloc << 1) + 0];
                        const float mk = (float)mask8[mloc];
                        atomicAdd(&sums[(size_t)(b * O_ + s) * 512 + nloc], v * mk);
                    } else if (nloc < 1024) {
                        outP[(size_t)mloc * 512 + (nloc - 512)] = v;
                    } else {
                        const int o = edges[((size_t)mloc << 1) + 1];
                        const float mk = (float)mask8[mloc];
                        atomicAdd(&sums[(size_t)(b * O_ + o) * 512 + (nloc - 1024)], v * mk);
                    }
                }
            }
        }
    }
}

__global__ void zero_f32(float* __restrict__ p, size_t n) {
    size_t i = (size_t)blockIdx.x * blockDim.x + threadIdx.x;
    if (i < n) p[i] = 0.0f;
}

__global__ void counts_kernel(const int* __restrict__ edges,
                              const int* __restrict__ ttype,
                              const int* __restrict__ pids,
                              const unsigned char* __restrict__ mask8,
                              const float* __restrict__ ptw,
                              float* __restrict__ counts) {
    int m = blockIdx.x * blockDim.x + threadIdx.x;   // 0..B*T-1
    if (m >= B_ * T_) return;
    const int tt = ttype[m];
    const float conf = (tt == 0) ? 1.0f : sigmoidf_(ptw[pids[m]]);
    const float w = conf * (float)mask8[m];
    const int b = m >> 10;
    const int s = edges[2 * m + 0];
    const int o = edges[2 * m + 1];
    atomicAdd(&counts[b * O_ + s], w);
    atomicAdd(&counts[b * O_ + o], w);
}

__global__ void normalize_kernel(float* __restrict__ sums,
                                 const float* __restrict__ counts) {
    size_t i = (size_t)blockIdx.x * blockDim.x + threadIdx.x;
    const size_t total = (size_t)B_ * O_ * 512;
    if (i >= total) return;
    const float c = counts[i >> 9];
    if (c > 0.0f) sums[i] /= c;
}

extern "C" void kernel_launch(void* const* d_in, const int* in_sizes, int n_in,
                              void* d_out, int out_size, void* d_ws, size_t ws_size,
                              hipStream_t stream) {
    const float* obj_vecs  = (const float*)d_in[0];   // (B, O, 512)
    const float* pred_vecs = (const float*)d_in[1];   // (B, T, 512)
    const float* W1a = (const float*)d_in[2];         // (1536, 512)
    const float* b1a = (const float*)d_in[3];         // (512)
    const float* W1b = (const float*)d_in[4];         // (512, 1536)
    const float* b1b = (const float*)d_in[5];         // (1536)
    const float* W2a = (const float*)d_in[6];         // (512, 512)
    const float* b2a = (const float*)d_in[7];
    const float* W2b = (const float*)d_in[8];         // (512, 512)
    const float* b2b = (const float*)d_in[9];
    const float* ptw = (const float*)d_in[10];        // (50)
    const int*   edges = (const int*)d_in[11];        // (B, T, 2)
    const unsigned char* mask8 = (const unsigned char*)d_in[12]; // bool (B,T)
    const int*   ttype = (const int*)d_in[13];        // (B, T)
    const int*   pids  = (const int*)d_in[14];        // (B, T)

    // outputs: [new_obj (B*O*512) | new_p (B*T*512)]
    float* out_obj = (float*)d_out;
    float* out_p   = (float*)d_out + (size_t)B_ * O_ * 512;

    // workspace layout (floats)
    float* h      = (float*)d_ws;                       // B*T*512  (64 MB)
    float* sums   = h + (size_t)B_ * T_ * 512;          // B*O*512  (16 MB)
    float* counts = sums + (size_t)B_ * O_ * 512;       // B*O
    float* h2     = counts + (size_t)B_ * O_;           // B*O*512  (16 MB)

    const int MT = B_ * T_;   // 32768
    const int MO = B_ * O_;   // 8192

    // 1) zero pooled sums + counts (contiguous)
    {
        size_t n = (size_t)MO * 512 + MO;
        zero_f32<<<dim3((unsigned)((n + 255) / 256)), dim3(256), 0, stream>>>(sums, n);
    }

    // 2) h = relu([obj[s]|pred|obj[o]] @ W1a + b1a)   M=32768 N=512 K=1536
    gemm_wmma_f32<1, 0><<<dim3(512 / BN, MT / BM), dim3(NTHREADS), 0, stream>>>(
        nullptr, obj_vecs, pred_vecs, edges, W1a, b1a,
        h, nullptr, nullptr, nullptr, nullptr, nullptr, nullptr,
        MT, 512, 3 * 512);

    // 3) new_t = relu(h @ W1b + b1b) * conf; split -> {sums, new_p, sums}
    gemm_wmma_f32<0, 1><<<dim3(1536 / BN, MT / BM), dim3(NTHREADS), 0, stream>>>(
        h, nullptr, nullptr, edges, W1b, b1b,
        nullptr, sums, out_p, ttype, pids, mask8, ptw,
        MT, 1536, 512);

    // 4) conf-weighted degree counts
    counts_kernel<<<dim3(MT / 256), dim3(256), 0, stream>>>(
        edges, ttype, pids, mask8, ptw, counts);

    // 5) pooled = counts>0 ? sums/counts : sums   (in place)
    normalize_kernel<<<dim3((unsigned)(((size_t)MO * 512) / 256)), dim3(256), 0, stream>>>(
        sums, counts);

    // 6) h2 = relu(pooled @ W2a + b2a)   M=8192 N=512 K=512
    gemm_wmma_f32<0, 0><<<dim3(512 / BN, MO / BM), dim3(NTHREADS), 0, stream>>>(
        sums, nullptr, nullptr, nullptr, W2a, b2a,
        h2, nullptr, nullptr, nullptr, nullptr, nullptr, nullptr,
        MO, 512, 512);

    // 7) new_obj = relu(h2 @ W2b + b2b)
    gemm_wmma_f32<0, 0><<<dim3(512 / BN, MO / BM), dim3(NTHREADS), 0, stream>>>(
        h2, nullptr, nullptr, nullptr, W2b, b2b,
        out_obj, nullptr, nullptr, nullptr, nullptr, nullptr, nullptr,
        MO, 512, 512);
}